// Classifier_22119081575034
// MI455X (gfx1250) — compile-verified
//
#include <hip/hip_runtime.h>

#define N_NODES 50000
#define N_EDGES 600000
#define N_RELS  19
#define D_IN    128
#define D_HID   64
#define D_OUT   2

typedef __attribute__((ext_vector_type(2))) float v2f;
typedef __attribute__((ext_vector_type(8))) float v8f;

__global__ void zero_f32(float* __restrict__ p, int n) {
    int i = blockIdx.x * blockDim.x + threadIdx.x;
    if (i < n) p[i] = 0.0f;
}

__global__ void hist_kernel(const int* __restrict__ etype, int* __restrict__ cnt) {
    int e = blockIdx.x * blockDim.x + threadIdx.x;
    if (e < N_EDGES) atomicAdd(&cnt[etype[e]], 1);
}

// Prefix sums: edge offsets, scatter cursors, and *block* offsets where each
// relation's 16-edge tiles are padded to a multiple of 8 (8 waves per block),
// so every block works on exactly one relation (uniform LDS staging).
__global__ void scan_kernel(const int* __restrict__ cnt, int* __restrict__ off,
                            int* __restrict__ cur, int* __restrict__ boff) {
    if (blockIdx.x == 0 && threadIdx.x == 0) {
        int o = 0, b = 0;
        for (int r = 0; r < N_RELS; ++r) {
            off[r] = o; cur[r] = o; boff[r] = b;
            o += cnt[r];
            int tiles = (cnt[r] + 15) >> 4;
            b += (tiles + 7) >> 3;
        }
        off[N_RELS] = o; boff[N_RELS] = b;
    }
}

__global__ void scatter_kernel(const int* __restrict__ src, const int* __restrict__ dst,
                               const int* __restrict__ etype, int* __restrict__ cur,
                               int* __restrict__ ssrc, int* __restrict__ sdst) {
    int e = blockIdx.x * blockDim.x + threadIdx.x;
    if (e < N_EDGES) {
        int pos = atomicAdd(&cur[etype[e]], 1);
        ssrc[pos] = src[e];
        sdst[pos] = dst[e];
    }
}

// Gathered GEMM per relation: agg[sdst[e]] += X[ssrc[e]] @ W[rel].
// One relation per block; W[rel] staged transposed+padded into LDS once,
// 8 waves x one 16-edge tile each; f32 WMMA 16x16 output, K=4 per step.
template <int K, int OUTC>
__global__ void rel_gemm_scatter(const float* __restrict__ X,    // [nodes, K]
                                 const float* __restrict__ W,    // [R, K, OUTC]
                                 const int* __restrict__ ssrc,
                                 const int* __restrict__ sdst,
                                 const int* __restrict__ off,    // [R+1] edge offsets
                                 const int* __restrict__ boff,   // [R+1] block offsets
                                 float* __restrict__ agg)        // [nodes, OUTC]
{
    constexpr int NT = (OUTC + 15) / 16;
    constexpr int KP = K + 2;                 // pad: bank-skew columns
    __shared__ float sW[OUTC * KP];           // W[rel] transposed: [col][k]

    int blk = blockIdx.x;
    if (blk >= boff[N_RELS]) return;          // uniform across block
    int r = 0;
    while (r < N_RELS - 1 && blk >= boff[r + 1]) ++r;

    const float* __restrict__ Wr = W + r * K * OUTC;
    for (int i = threadIdx.x; i < K * OUTC; i += blockDim.x) {
        int d = i / OUTC, c = i % OUTC;       // coalesced read, conflict-free write
        sW[c * KP + d] = Wr[i];
    }
    __syncthreads();

    int tiles_r = (off[r + 1] - off[r] + 15) >> 4;
    int tile = ((blk - boff[r]) << 3) + (threadIdx.x >> 5);
    if (tile >= tiles_r) return;              // wave-uniform after barrier

    int es = off[r] + (tile << 4);
    int ee = min(es + 16, off[r + 1]);
    int lane = threadIdx.x & 31;
    int half = lane >> 4;                     // lanes 16-31 hold K+2,K+3
    int l = lane & 15;
    const float* __restrict__ Arow = X + ssrc[min(es + l, ee - 1)] * K;

    v8f acc[NT] = {};
    for (int k0 = 0; k0 < K; k0 += 4) {
        int ka = k0 + 2 * half;
        v2f a = *(const v2f*)(Arow + ka);     // global_load_b64
#pragma unroll
        for (int nt = 0; nt < NT; ++nt) {
            int col = nt * 16 + l;
            v2f b;
            if (col < OUTC) b = *(const v2f*)(sW + col * KP + ka);  // ds_load_b64
            else            { b.x = 0.0f; b.y = 0.0f; }
            acc[nt] = __builtin_amdgcn_wmma_f32_16x16x4_f32(
                false, a, false, b, (short)0, acc[nt], false, false);
        }
    }

    // branch-free clamped loads; only the atomics are predicated
    int dn[8];
    bool dv[8];
#pragma unroll
    for (int j = 0; j < 8; ++j) {             // D VGPR j -> tile rows j / j+8
        int e = es + j + 8 * half;
        dv[j] = (e < ee);
        dn[j] = sdst[min(e, ee - 1)];
    }
#pragma unroll
    for (int nt = 0; nt < NT; ++nt) {
        int col = nt * 16 + l;
        if (col < OUTC) {
#pragma unroll
            for (int j = 0; j < 8; ++j)
                if (dv[j]) atomicAdd(&agg[dn[j] * OUTC + col], acc[nt][j]);
        }
    }
}

// Dense self-loop GEMM with fused epilogue: out = act(agg + X @ Wl + bias).
// Wl staged transposed in LDS; 8 waves per block, one 16-node tile per wave.
template <int K, int OUTC, bool RELU>
__global__ void dense_gemm_epilogue(const float* __restrict__ X,    // [n, K]
                                    const float* __restrict__ Wl,   // [K, OUTC]
                                    const float* __restrict__ bias, // [OUTC]
                                    const float* __restrict__ agg,  // [n, OUTC]
                                    float* __restrict__ out,        // [n, OUTC]
                                    int n)
{
    constexpr int NT = (OUTC + 15) / 16;
    constexpr int KP = K + 2;
    __shared__ float sW[OUTC * KP];

    for (int i = threadIdx.x; i < K * OUTC; i += blockDim.x) {
        int d = i / OUTC, c = i % OUTC;
        sW[c * KP + d] = Wl[i];
    }
    __syncthreads();

    int tile = (blockIdx.x << 3) + (threadIdx.x >> 5);
    int nb = tile << 4;
    if (nb >= n) return;
    int lane = threadIdx.x & 31;
    int half = lane >> 4;
    int l = lane & 15;
    const float* __restrict__ Arow = X + min(nb + l, n - 1) * K;

    v8f acc[NT] = {};
    for (int k0 = 0; k0 < K; k0 += 4) {
        int ka = k0 + 2 * half;
        v2f a = *(const v2f*)(Arow + ka);
#pragma unroll
        for (int nt = 0; nt < NT; ++nt) {
            int col = nt * 16 + l;
            v2f b;
            if (col < OUTC) b = *(const v2f*)(sW + col * KP + ka);
            else            { b.x = 0.0f; b.y = 0.0f; }
            acc[nt] = __builtin_amdgcn_wmma_f32_16x16x4_f32(
                false, a, false, b, (short)0, acc[nt], false, false);
        }
    }
#pragma unroll
    for (int nt = 0; nt < NT; ++nt) {
        int col = nt * 16 + l;
        if (col < OUTC) {
#pragma unroll
            for (int j = 0; j < 8; ++j) {
                int node = nb + j + 8 * half;
                if (node < n) {
                    float v = acc[nt][j] + agg[node * OUTC + col] + bias[col];
                    if (RELU) v = fmaxf(v, 0.0f);
                    out[node * OUTC + col] = v;
                }
            }
        }
    }
}

extern "C" void kernel_launch(void* const* d_in, const int* in_sizes, int n_in,
                              void* d_out, int out_size, void* d_ws, size_t ws_size,
                              hipStream_t stream) {
    (void)in_sizes; (void)n_in; (void)out_size; (void)ws_size;
    const float* feat  = (const float*)d_in[0];
    const float* W1    = (const float*)d_in[1];
    const float* loop1 = (const float*)d_in[2];
    const float* b1    = (const float*)d_in[3];
    const float* W2    = (const float*)d_in[4];
    const float* loop2 = (const float*)d_in[5];
    const float* b2    = (const float*)d_in[6];
    const int*   src   = (const int*)d_in[7];
    const int*   dst   = (const int*)d_in[8];
    const int*   etype = (const int*)d_in[9];
    float* out = (float*)d_out;

    char* ws = (char*)d_ws;
    size_t o = 0;
    auto alloc = [&](size_t bytes) -> void* {
        void* p = ws + o;
        o = (o + bytes + 255) & ~(size_t)255;
        return p;
    };
    float* agg1 = (float*)alloc(sizeof(float) * N_NODES * D_HID);
    float* agg2 = (float*)alloc(sizeof(float) * N_NODES * D_OUT);
    float* h1   = (float*)alloc(sizeof(float) * N_NODES * D_HID);
    int* ssrc = (int*)alloc(sizeof(int) * N_EDGES);
    int* sdst = (int*)alloc(sizeof(int) * N_EDGES);
    int* cnt  = (int*)alloc(sizeof(int) * N_RELS);
    int* cur  = (int*)alloc(sizeof(int) * N_RELS);
    int* off  = (int*)alloc(sizeof(int) * (N_RELS + 1));
    int* boff = (int*)alloc(sizeof(int) * (N_RELS + 1));

    const int T = 256;

    // 1) zero accumulators + histogram (every call: deterministic)
    zero_f32<<<(N_NODES * D_HID + T - 1) / T, T, 0, stream>>>(agg1, N_NODES * D_HID);
    zero_f32<<<(N_NODES * D_OUT + T - 1) / T, T, 0, stream>>>(agg2, N_NODES * D_OUT);
    zero_f32<<<1, 32, 0, stream>>>((float*)cnt, N_RELS);

    // 2) counting sort of edges by relation
    hist_kernel<<<(N_EDGES + T - 1) / T, T, 0, stream>>>(etype, cnt);
    scan_kernel<<<1, 1, 0, stream>>>(cnt, off, cur, boff);
    scatter_kernel<<<(N_EDGES + T - 1) / T, T, 0, stream>>>(src, dst, etype, cur, ssrc, sdst);

    // 3) layer 1: relational messages (WMMA + atomic scatter), then self-loop + ReLU
    int max_blocks = N_EDGES / 128 + N_RELS + 1;   // >= sum ceil(tiles_r/8)
    rel_gemm_scatter<D_IN, D_HID><<<max_blocks, T, 0, stream>>>(
        feat, W1, ssrc, sdst, off, boff, agg1);
    int dense_blocks = ((N_NODES + 15) / 16 + 7) / 8;
    dense_gemm_epilogue<D_IN, D_HID, true><<<dense_blocks, T, 0, stream>>>(
        feat, loop1, b1, agg1, h1, N_NODES);

    // 4) layer 2: same pipeline, K=64 -> 2 outputs (B cols >=2 zero-padded)
    rel_gemm_scatter<D_HID, D_OUT><<<max_blocks, T, 0, stream>>>(
        h1, W2, ssrc, sdst, off, boff, agg2);
    dense_gemm_epilogue<D_HID, D_OUT, false><<<dense_blocks, T, 0, stream>>>(
        h1, loop2, b2, agg2, out, N_NODES);
}